// HardEMQuantizer_77068893160052
// MI455X (gfx1250) — compile-verified
//
#include <hip/hip_runtime.h>
#include <hip/hip_bf16.h>
#include <stdint.h>

// Problem constants (from reference): B=8, T=2048, M=8, K=512, D=64
#define MM    8
#define KK    512
#define DD    64
#define NTOK  (8 * 2048)            // 16384 tokens
#define PAIRS (NTOK * MM)           // 131072 (n,m) pairs
#define QSIZE ((size_t)NTOK * MM * DD)  // 8388608 floats per output copy

#define WPB   8                     // waves per block (256 threads, wave32)
#define NBLK  2048
#define TOTW  (WPB * NBLK)          // 16384 waves
#define PPW   (PAIRS / TOTW)        // 8 pairs per wave

#if defined(__has_builtin)
#if __has_builtin(__builtin_amdgcn_global_load_async_to_lds_b128)
#define HAVE_ASYNC 1
#endif
#if __has_builtin(__builtin_amdgcn_s_wait_asynccnt)
#define HAVE_WAITASYNC 1
#endif
#endif

#if defined(HAVE_WAITASYNC)
#define WAIT_ASYNC(n) do { __builtin_amdgcn_s_wait_asynccnt(n); asm volatile("" ::: "memory"); } while (0)
#else
#define WAIT_ASYNC(n) asm volatile("s_wait_asynccnt %0" :: "i"(n) : "memory")
#endif

// The builtin expects pointers to 16-byte int vectors in explicit address spaces
// (per hipcc diagnostic: param type is `vector_size(16) int __device__ *`).
typedef int v4i __attribute__((vector_size(16)));
typedef __attribute__((address_space(1))) v4i as1_v4i;   // global
typedef __attribute__((address_space(3))) v4i as3_v4i;   // LDS

// Copy one 512-float (2KB) logits tile into LDS with 4 async B128 transfers.
// IOFFSET (imm) is added to BOTH the global and the LDS address (ISA 08 §4.4),
// and must be an integer constant expression -> literal offsets.
__device__ __forceinline__ void async_tile(const float* gsrc, float* ldst, int lane) {
#if defined(HAVE_ASYNC)
  as1_v4i* g = (as1_v4i*)(v4i*)(const_cast<float*>(gsrc) + lane * 4);  // 16 B per lane
  as3_v4i* l = (as3_v4i*)(v4i*)(ldst + lane * 4);
  __builtin_amdgcn_global_load_async_to_lds_b128(g, l,    0, 0);
  __builtin_amdgcn_global_load_async_to_lds_b128(g, l,  512, 0);
  __builtin_amdgcn_global_load_async_to_lds_b128(g, l, 1024, 0);
  __builtin_amdgcn_global_load_async_to_lds_b128(g, l, 1536, 0);
#endif
}

__global__ __launch_bounds__(256)
void hardem_quantizer_kernel(const float* __restrict__ logits,
                             const float* __restrict__ emb,
                             float* __restrict__ out) {
  const int lane  = threadIdx.x & 31;
  const int lwave = threadIdx.x >> 5;
  const int wave  = blockIdx.x * WPB + lwave;
  const int pair0 = wave * PPW;
  if (pair0 >= PAIRS) return;

#if defined(HAVE_ASYNC)
  __shared__ __align__(16) float lds[WPB * 2 * KK];   // 32 KB: 2x2KB per wave
  float* buf = &lds[lwave * 2 * KK];
  // Prefetch tile 0
  async_tile(logits + (size_t)pair0 * KK, buf, lane);
#endif

  for (int i = 0; i < PPW; ++i) {
    const int pair = pair0 + i;

#if defined(HAVE_ASYNC)
    if (i + 1 < PPW) {
      // Kick off next tile, then wait for current (in-order completion:
      // asynccnt<=4 means the 4 older transfers have landed in LDS).
      async_tile(logits + (size_t)(pair + 1) * KK, buf + ((i + 1) & 1) * KK, lane);
      WAIT_ASYNC(4);
    } else {
      WAIT_ASYNC(0);
    }
    const float* src = buf + (i & 1) * KK;
#else
    const float* src = logits + (size_t)pair * KK;
#endif

    // ---- argmax over K=512: 16 candidates per lane, then wave reduction ----
    float bestV = -3.402823466e+38f;
    int   bestK = 0x7fffffff;
#pragma unroll
    for (int jj = 0; jj < 4; ++jj) {
      const float4 v = *(const float4*)(src + jj * 128 + lane * 4);
      const int kb = jj * 128 + lane * 4;
      if (v.x > bestV || (v.x == bestV && kb + 0 < bestK)) { bestV = v.x; bestK = kb + 0; }
      if (v.y > bestV || (v.y == bestV && kb + 1 < bestK)) { bestV = v.y; bestK = kb + 1; }
      if (v.z > bestV || (v.z == bestV && kb + 2 < bestK)) { bestV = v.z; bestK = kb + 2; }
      if (v.w > bestV || (v.w == bestV && kb + 3 < bestK)) { bestV = v.w; bestK = kb + 3; }
    }
#pragma unroll
    for (int s = 16; s >= 1; s >>= 1) {
      const float ov = __shfl_xor(bestV, s, 32);
      const int   ok = __shfl_xor(bestK, s, 32);
      // first-occurrence tie-break == jnp.argmax semantics
      if (ov > bestV || (ov == bestV && ok < bestK)) { bestV = ov; bestK = ok; }
    }

    // ---- gather embeddings[m, bestK, :] and write both output copies ----
    const int m = pair & (MM - 1);
    const float2 e = *(const float2*)(emb + (size_t)(m * KK + bestK) * DD + lane * 2);
    const size_t qoff = (size_t)pair * DD + lane * 2;   // (n*M + m)*D
    *(float2*)(out + qoff)         = e;   // quantized (B,T,M*D)
    *(float2*)(out + QSIZE + qoff) = e;   // quantized_stack (B,T,M,D) — same flat layout
    if (lane == 0) {
      ((int*)(out + 2 * QSIZE))[pair] = bestK;          // encoding_indices int32
    }
  }
}

extern "C" void kernel_launch(void* const* d_in, const int* in_sizes, int n_in,
                              void* d_out, int out_size, void* d_ws, size_t ws_size,
                              hipStream_t stream) {
  const float* logits = (const float*)d_in[0];  // (8, 2048, 4096) f32
  const float* emb    = (const float*)d_in[1];  // (8, 512, 64)  f32
  float* out = (float*)d_out;                   // quantized ++ quantized_stack ++ indices
  hardem_quantizer_kernel<<<NBLK, 256, 0, stream>>>(logits, emb, out);
}